// SimVQ_81690277970403
// MI455X (gfx1250) — compile-verified
//
#include <hip/hip_runtime.h>
#include <hip/hip_bf16.h>

// ---------------------------------------------------------------------------
// SimVQ on MI455X (gfx1250, wave32).  fp32 WMMA (v_wmma_f32_16x16x4_f32) for
// both GEMMs: imp = codebook @ W^T   and the NN-search GEMM x @ imp^T.
// ---------------------------------------------------------------------------

typedef float v2f __attribute__((ext_vector_type(2)));
typedef float v8f __attribute__((ext_vector_type(8)));

__device__ __forceinline__ v8f wmma_f32(v2f a, v2f b, v8f c) {
    // 8 args: (neg_a, A, neg_b, B, c_mod, C, reuse_a, reuse_b)
    return __builtin_amdgcn_wmma_f32_16x16x4_f32(false, a, false, b,
                                                 (short)0, c, false, false);
}

#define KP   64          // K panel staged in LDS per step
#define LST  68          // LDS row stride (floats): 68 = 64+4 -> conflict-free, 16B aligned

// ---------------------------------------------------------------------------
// Kernel 1: imp[c, d] = sum_k codebook[c,k] * W[d,k]     (M=C, N=D, K=D)
// Block = 128 threads (4 waves), tile 64(M) x 64(N); wave wv owns rows wv*16..+15.
// ---------------------------------------------------------------------------
__global__ __launch_bounds__(128) void simvq_imp(const float* __restrict__ cb,
                                                 const float* __restrict__ W,
                                                 float* __restrict__ imp,
                                                 int C, int D) {
    __shared__ float as[64 * LST];   // codebook panel  [m][k]
    __shared__ float bs[64 * LST];   // W panel         [n][k]  (B[k][n] = W[n][k])
    const int tid  = threadIdx.x;
    const int lane = tid & 31, wv = tid >> 5;
    const int mn   = lane & 15, half = lane >> 4;
    const int rowBase = blockIdx.y * 64;   // codebook rows
    const int colBase = blockIdx.x * 64;   // output cols = W rows

    v8f acc[4];
#pragma unroll
    for (int t = 0; t < 4; ++t)
#pragma unroll
        for (int j = 0; j < 8; ++j) acc[t][j] = 0.0f;

    for (int ko = 0; ko < D; ko += KP) {
        __syncthreads();
        for (int i = tid; i < 64 * (KP / 4); i += 128) {
            const int r  = i >> 4;           // KP/4 = 16 float4 per row
            const int c4 = (i & 15) * 4;
            *(float4*)&as[r * LST + c4] =
                *(const float4*)&cb[(size_t)(rowBase + r) * D + ko + c4];
            *(float4*)&bs[r * LST + c4] =
                *(const float4*)&W[(size_t)(colBase + r) * D + ko + c4];
        }
        __syncthreads();
#pragma unroll 4
        for (int kl = 0; kl < KP; kl += 4) {
            const v2f a = *(const v2f*)&as[(wv * 16 + mn) * LST + kl + 2 * half];
#pragma unroll
            for (int t = 0; t < 4; ++t) {
                const v2f b = *(const v2f*)&bs[(t * 16 + mn) * LST + kl + 2 * half];
                acc[t] = wmma_f32(a, b, acc[t]);
            }
        }
    }
#pragma unroll
    for (int t = 0; t < 4; ++t)
#pragma unroll
        for (int j = 0; j < 8; ++j) {
            const int r = rowBase + wv * 16 + j + 8 * half;
            const int c = colBase + t * 16 + mn;
            imp[(size_t)r * D + c] = acc[t][j];
        }
}

// ---------------------------------------------------------------------------
// Kernel 2: inorm2[c] = sum_k imp[c,k]^2   (wave per row)
// ---------------------------------------------------------------------------
__global__ __launch_bounds__(256) void simvq_inorm(const float* __restrict__ imp,
                                                   float* __restrict__ inorm2,
                                                   int D) {
    const int lane = threadIdx.x & 31, wv = threadIdx.x >> 5;
    const int row  = blockIdx.x * 8 + wv;
    float s = 0.f;
    for (int i = lane; i < D; i += 32) {
        const float v = imp[(size_t)row * D + i];
        s += v * v;
    }
#pragma unroll
    for (int m = 16; m >= 1; m >>= 1) s += __shfl_xor(s, m, 32);
    if (lane == 0) inorm2[row] = s;
}

// ---------------------------------------------------------------------------
// Kernel 3: nearest neighbor.  Block = 128 threads (4 waves) owns 64 x-rows,
// sweeps all C codes in 64-col chunks.  score = inorm2[c] - 2 * (x . imp_c).
// ---------------------------------------------------------------------------
__global__ __launch_bounds__(128) void simvq_nn(const float* __restrict__ xf,
                                                const float* __restrict__ imp,
                                                const float* __restrict__ inorm2,
                                                int* __restrict__ idxOut,
                                                float* __restrict__ idxOutF,
                                                int C, int D) {
    __shared__ float xs[64 * LST];
    __shared__ float cs[64 * LST];
    const int tid  = threadIdx.x;
    const int lane = tid & 31, wv = tid >> 5;
    const int mn   = lane & 15, half = lane >> 4;
    const int rowBase = blockIdx.x * 64;

    float bestV[8];
    int   bestI[8];
#pragma unroll
    for (int j = 0; j < 8; ++j) { bestV[j] = 3.402823e38f; bestI[j] = 0; }

    const int nChunks = C / 64;
    for (int ch = 0; ch < nChunks; ++ch) {
        const int colBase = ch * 64;
        v8f acc[4];
#pragma unroll
        for (int t = 0; t < 4; ++t)
#pragma unroll
            for (int j = 0; j < 8; ++j) acc[t][j] = 0.0f;

        for (int ko = 0; ko < D; ko += KP) {
            __syncthreads();
            for (int i = tid; i < 64 * (KP / 4); i += 128) {
                const int r  = i >> 4;
                const int c4 = (i & 15) * 4;
                *(float4*)&xs[r * LST + c4] =
                    *(const float4*)&xf[(size_t)(rowBase + r) * D + ko + c4];
                *(float4*)&cs[r * LST + c4] =
                    *(const float4*)&imp[(size_t)(colBase + r) * D + ko + c4];
            }
            __syncthreads();
#pragma unroll 4
            for (int kl = 0; kl < KP; kl += 4) {
                const v2f a = *(const v2f*)&xs[(wv * 16 + mn) * LST + kl + 2 * half];
#pragma unroll
                for (int t = 0; t < 4; ++t) {
                    const v2f b = *(const v2f*)&cs[(t * 16 + mn) * LST + kl + 2 * half];
                    acc[t] = wmma_f32(a, b, acc[t]);
                }
            }
        }
        // running per-row argmin (this lane covers cols c = colBase + t*16 + mn)
#pragma unroll
        for (int t = 0; t < 4; ++t) {
            const int c    = colBase + t * 16 + mn;
            const float q2 = inorm2[c];
#pragma unroll
            for (int j = 0; j < 8; ++j) {
                const float v = q2 - 2.0f * acc[t][j];
                if (v < bestV[j]) { bestV[j] = v; bestI[j] = c; }
            }
        }
    }

    // reduce across the 16 lanes sharing the same rows (tie-break: first index)
#pragma unroll
    for (int m = 1; m < 16; m <<= 1) {
#pragma unroll
        for (int j = 0; j < 8; ++j) {
            const float ov = __shfl_xor(bestV[j], m, 16);
            const int   oi = __shfl_xor(bestI[j], m, 16);
            if (ov < bestV[j] || (ov == bestV[j] && oi < bestI[j])) {
                bestV[j] = ov; bestI[j] = oi;
            }
        }
    }
    if (mn == 0) {
#pragma unroll
        for (int j = 0; j < 8; ++j) {
            const int row = rowBase + wv * 16 + j + 8 * half;
            idxOut[row]  = bestI[j];
            idxOutF[row] = (float)bestI[j];
        }
    }
}

// ---------------------------------------------------------------------------
// Kernel 4: gather + rotation trick + commit-loss row sums.  Wave per row.
// ---------------------------------------------------------------------------
__global__ __launch_bounds__(256) void simvq_rot(const float* __restrict__ xf,
                                                 const float* __restrict__ imp,
                                                 const int* __restrict__ idxws,
                                                 float* __restrict__ outQ,
                                                 float* __restrict__ rowsum,
                                                 int D) {
    const int lane = threadIdx.x & 31, wv = threadIdx.x >> 5;
    const int t = blockIdx.x * 8 + wv;
    const float* xr = &xf[(size_t)t * D];
    const int idx   = idxws[t];
    const float* qr = &imp[(size_t)idx * D];

    float xv[16], qv[16], sv[16];
    float nx2 = 0.f, nq2 = 0.f, csum = 0.f;
#pragma unroll
    for (int i = 0; i < 16; ++i) {
        const float xx = xr[i * 32 + lane];
        const float qq = qr[i * 32 + lane];
        xv[i] = xx; qv[i] = qq;
        nx2 += xx * xx;
        nq2 += qq * qq;
        const float d = xx - qq + 1e-6f;
        csum += d * d;
    }
#pragma unroll
    for (int m = 16; m >= 1; m >>= 1) {
        nx2  += __shfl_xor(nx2, m, 32);
        nq2  += __shfl_xor(nq2, m, 32);
        csum += __shfl_xor(csum, m, 32);
    }
    const float normx = sqrtf(nx2), normq = sqrtf(nq2);
    const float dx = fmaxf(normx, 1e-6f), dq = fmaxf(normq, 1e-6f);

    float ns2 = 0.f;
#pragma unroll
    for (int i = 0; i < 16; ++i) {
        const float s = xv[i] / dx + qv[i] / dq;   // u + qhat
        sv[i] = s;
        ns2 += s * s;
    }
#pragma unroll
    for (int m = 16; m >= 1; m >>= 1) ns2 += __shfl_xor(ns2, m, 32);
    const float ns = fmaxf(sqrtf(ns2), 1e-12f);

    float ew = 0.f;
#pragma unroll
    for (int i = 0; i < 16; ++i) ew += xv[i] * (sv[i] / ns);
#pragma unroll
    for (int m = 16; m >= 1; m >>= 1) ew += __shfl_xor(ew, m, 32);

    const float eu    = nx2 / dx;        // sum x_i * (x_i / dx)
    const float scale = normq / dx;
#pragma unroll
    for (int i = 0; i < 16; ++i) {
        const float w  = sv[i] / ns;
        const float qh = qv[i] / dq;
        const float o  = (xv[i] - 2.0f * ew * w + 2.0f * eu * qh) * scale;
        outQ[(size_t)t * D + i * 32 + lane] = o;
    }
    if (lane == 0) rowsum[t] = csum;
}

// ---------------------------------------------------------------------------
// Kernel 5: commit_loss = mean(rowsum)
// ---------------------------------------------------------------------------
__global__ __launch_bounds__(256) void simvq_loss(const float* __restrict__ rowsum,
                                                  float* __restrict__ outLoss,
                                                  int T) {
    __shared__ float red[8];
    float s = 0.f;
    for (int i = threadIdx.x; i < T; i += 256) s += rowsum[i];
#pragma unroll
    for (int m = 16; m >= 1; m >>= 1) s += __shfl_xor(s, m, 32);
    const int lane = threadIdx.x & 31, wv = threadIdx.x >> 5;
    if (lane == 0) red[wv] = s;
    __syncthreads();
    if (threadIdx.x == 0) {
        float tot = 0.f;
        for (int i = 0; i < 8; ++i) tot += red[i];
        outLoss[0] = tot / (float)T;
    }
}

// ---------------------------------------------------------------------------
extern "C" void kernel_launch(void* const* d_in, const int* in_sizes, int n_in,
                              void* d_out, int out_size, void* d_ws, size_t ws_size,
                              hipStream_t stream) {
    (void)n_in; (void)out_size; (void)ws_size;
    const float* x  = (const float*)d_in[0];   // [T, D]
    const float* W  = (const float*)d_in[1];   // [D, D]
    const float* cb = (const float*)d_in[2];   // [C, D]

    int D = 1;
    while ((long long)D * D < (long long)in_sizes[1]) ++D;   // D = 512
    const int T = in_sizes[0] / D;                            // 16384
    const int C = in_sizes[2] / D;                            // 8192

    // workspace layout
    float* imp    = (float*)d_ws;                 // C*D
    float* inorm2 = imp + (size_t)C * D;          // C
    int*   idxws  = (int*)(inorm2 + C);           // T
    float* rowsum = (float*)(idxws + T);          // T

    // output layout: quantized [T*D] | indices (as float) [T] | commit_loss [1]
    float* outQ    = (float*)d_out;
    float* outIdxF = outQ + (size_t)T * D;
    float* outLoss = outIdxF + T;

    simvq_imp<<<dim3(D / 64, C / 64), 128, 0, stream>>>(cb, W, imp, C, D);
    simvq_inorm<<<C / 8, 256, 0, stream>>>(imp, inorm2, D);
    simvq_nn<<<T / 64, 128, 0, stream>>>(x, imp, inorm2, idxws, outIdxF, C, D);
    simvq_rot<<<T / 8, 256, 0, stream>>>(x, imp, idxws, outQ, rowsum, D);
    simvq_loss<<<1, 256, 0, stream>>>(rowsum, outLoss, T);
}